// HybridAttentionModule_46557445489505
// MI455X (gfx1250) — compile-verified
//
#include <hip/hip_runtime.h>
#include <math.h>

// Problem constants (match reference)
#define B_    16
#define C_    512
#define L_    4096
#define G_    32
#define CPG_  16
#define HALF_ 8
#define EPS_  1e-5f

#define RS_      (L_ + 4)     // padded LDS row stride in floats (bank-conflict-free column reads, 16B aligned)
#define NTHREADS 1024
#define NWAVES   (NTHREADS / 32)

// ---- CDNA5 async global->LDS copy (ASYNCcnt-tracked) ----------------------
typedef int v4i_ __attribute__((vector_size(4 * sizeof(int))));
#define AS1_ __attribute__((address_space(1)))
#define AS3_ __attribute__((address_space(3)))

#if __has_builtin(__builtin_amdgcn_global_load_async_to_lds_b128)
#define ASYNC_LOAD_B128(gp, lp)                                               \
    __builtin_amdgcn_global_load_async_to_lds_b128(                           \
        (AS1_ v4i_*)(gp), (AS3_ v4i_*)(lp), 0, 0)
#else
// Fallback: raw gfx1250 encoding — vdst = LDS byte offset, vaddr = 64b global
#define ASYNC_LOAD_B128(gp, lp)                                               \
    asm volatile("global_load_async_to_lds_b128 %0, %1, off"                  \
                 :: "v"((unsigned)(size_t)(lp)), "v"(gp) : "memory")
#endif

static __device__ __forceinline__ void wait_async_zero() {
#if __has_builtin(__builtin_amdgcn_s_wait_asynccnt)
    __builtin_amdgcn_s_wait_asynccnt(0);
#else
    asm volatile("s_wait_asynccnt 0" ::: "memory");
#endif
}

static __device__ __forceinline__ float waveReduceAdd(float v) {
#pragma unroll
    for (int off = 16; off >= 1; off >>= 1)
        v += __shfl_xor(v, off, 32);
    return v;
}

__global__ __launch_bounds__(NTHREADS, 1)
void ham_fused_kernel(const float* __restrict__ x,
                      const float* __restrict__ cw_w, const float* __restrict__ cw_b,
                      const float* __restrict__ sw_w, const float* __restrict__ sw_b,
                      const float* __restrict__ gn_w, const float* __restrict__ gn_b,
                      float* __restrict__ out)
{
    __shared__ float tile[CPG_ * RS_];       // ~256.25 KB of the 320 KB WGP LDS
    __shared__ float s_p1[NWAVES];
    __shared__ float s_p2[NWAVES];
    __shared__ float s_p3[NWAVES];
    __shared__ float s_ca[HALF_];
    __shared__ float s_stats[2];

    const int tid = (int)threadIdx.x;
    const int bg  = (int)blockIdx.x;         // b*G + g
    const int g   = bg & (G_ - 1);
    const int b   = bg >> 5;                 // bg / G

    const float* src = x + (size_t)bg * CPG_ * L_;

    // ---- Phase 1: async-copy the 16 x 4096 fp32 tile into LDS -------------
    // 16384 b128 transfers, 16 per thread; all lanes active (EXEC all-1s).
#pragma unroll
    for (int k = 0; k < (CPG_ * (L_ / 4)) / NTHREADS; ++k) {
        int i   = tid + k * NTHREADS;
        int row = i >> 10;                   // 1024 float4 per row
        int c4  = (i & 1023) << 2;           // float offset within row
        const float* gp = src + row * L_ + c4;
        float*       lp = &tile[row * RS_ + c4];
        ASYNC_LOAD_B128(gp, lp);
    }
    wait_async_zero();                        // my wave's ASYNCcnt == 0
    __syncthreads();                          // all waves' LDS writes visible

    // ---- Phase 2: reductions (deterministic tree, no float atomics) -------
    // 128 threads per row r (0..7): row sums for x1, sum/sumsq for x2 row 8+r.
    {
        const int r = tid >> 7;               // 0..7
        const int j = tid & 127;
        float p1 = 0.f, p2 = 0.f, p3 = 0.f;
#pragma unroll 4
        for (int i = 0; i < L_; i += 128) {
            p1 += tile[r * RS_ + j + i];
            float v = tile[(HALF_ + r) * RS_ + j + i];
            p2 += v;
            p3 += v * v;
        }
        p1 = waveReduceAdd(p1);
        p2 = waveReduceAdd(p2);
        p3 = waveReduceAdd(p3);
        const int wid = tid >> 5;
        if ((tid & 31) == 0) { s_p1[wid] = p1; s_p2[wid] = p2; s_p3[wid] = p3; }
    }
    __syncthreads();
    if (tid < HALF_) {                        // 4 waves per row, fixed order
        s_ca[tid] = ((s_p1[4 * tid + 0] + s_p1[4 * tid + 1]) +
                     (s_p1[4 * tid + 2] + s_p1[4 * tid + 3]));
    }
    if (tid == 0) {
        float s = 0.f, ss = 0.f;
#pragma unroll
        for (int w = 0; w < NWAVES; ++w) { s += s_p2[w]; ss += s_p3[w]; }
        s_stats[0] = s; s_stats[1] = ss;
    }
    __syncthreads();

    // ---- Broadcast scalars: gates, norm coefficients -----------------------
    const float invL  = 1.0f / (float)L_;
    const float invN2 = 1.0f / (float)(HALF_ * L_);
    const float mu  = s_stats[0] * invN2;
    const float var = s_stats[1] * invN2 - mu * mu;
    const float rs  = rsqrtf(var + EPS_);
    const float cwwg = cw_w[g], cwbg = cw_b[g];
    const float swwg = sw_w[g], swbg = sw_b[g];

    float w1v[HALF_], a[HALF_];
    float d = 0.f;
#pragma unroll
    for (int h = 0; h < HALF_; ++h) {
        float ca = s_ca[h] * invL;
        w1v[h] = 1.0f / (1.0f + expf(-(ca * cwwg + cwbg)));
        float gw = gn_w[g * HALF_ + h];
        float gb = gn_b[g * HALF_ + h];
        a[h] = rs * gw * (1.0f / (float)HALF_);
        d   += (gb - mu * rs * gw) * (1.0f / (float)HALF_);
    }

    // ---- Phase 3: per-l gate + permuted, non-temporal writeout -------------
    // out[b, cp*G + g, l] = tile[cp][l] * (cp<8 ? w1[cp] : w2[l])
    float* dst = out + (size_t)b * C_ * L_ + (size_t)g * L_;
#pragma unroll
    for (int k = 0; k < L_ / NTHREADS; ++k) {
        int l = tid + k * NTHREADS;
        float acc = d;
#pragma unroll
        for (int h = 0; h < HALF_; ++h)
            acc += a[h] * tile[(HALF_ + h) * RS_ + l];   // banks differ by 4 per row: conflict-free
        float w2 = 1.0f / (1.0f + expf(-(acc * swwg + swbg)));
#pragma unroll
        for (int cp = 0; cp < HALF_; ++cp)
            __builtin_nontemporal_store(tile[cp * RS_ + l] * w1v[cp],
                                        dst + (size_t)(cp * G_) * L_ + l);
#pragma unroll
        for (int cp = HALF_; cp < CPG_; ++cp)
            __builtin_nontemporal_store(tile[cp * RS_ + l] * w2,
                                        dst + (size_t)(cp * G_) * L_ + l);
    }
}

extern "C" void kernel_launch(void* const* d_in, const int* in_sizes, int n_in,
                              void* d_out, int out_size, void* d_ws, size_t ws_size,
                              hipStream_t stream) {
    (void)in_sizes; (void)n_in; (void)out_size; (void)d_ws; (void)ws_size;
    const float* x    = (const float*)d_in[0];
    const float* cw_w = (const float*)d_in[1];
    const float* cw_b = (const float*)d_in[2];
    const float* sw_w = (const float*)d_in[3];
    const float* sw_b = (const float*)d_in[4];
    const float* gn_w = (const float*)d_in[5];
    const float* gn_b = (const float*)d_in[6];
    float* out = (float*)d_out;

    ham_fused_kernel<<<B_ * G_, NTHREADS, 0, stream>>>(
        x, cw_w, cw_b, sw_w, sw_b, gn_w, gn_b, out);
}